// EinsteinPINN_84164179133338
// MI455X (gfx1250) — compile-verified
//
#include <hip/hip_runtime.h>
#include <cstddef>
#include <cstdint>

#ifndef __has_builtin
#define __has_builtin(x) 0
#endif

#define NB 8192
#define H  128

typedef __attribute__((ext_vector_type(2))) float        v2f;
typedef __attribute__((ext_vector_type(8))) float        v8f;
typedef __attribute__((ext_vector_type(4))) unsigned int v4u;
typedef __attribute__((ext_vector_type(4))) int          v4i;
typedef __attribute__((ext_vector_type(8))) int          v8i;

// gfx1250 has a hardware tanh transcendental; fall back to a branchless
// exp2/rcp formulation (tanh = (t-1)/(t+1), t = 2^(2x*log2 e)) if the builtin
// is not declared by this toolchain.
__device__ __forceinline__ float fast_tanh(float z) {
#if __has_builtin(__builtin_amdgcn_tanhf)
  return __builtin_amdgcn_tanhf(z);
#else
  float t = __builtin_amdgcn_exp2f(z * 2.8853900817779268f); // 2*log2(e)
  float r = __builtin_amdgcn_rcpf(t + 1.0f);
  return 1.0f - 2.0f * r;
#endif
}

// broadcast from the low half-wave lane (lane & 15): gives all lanes the value
// held by the lane that owns channels 0..7 of the same feature column.
__device__ __forceinline__ float bc16(float v) {
  return __shfl(v, (int)(threadIdx.x & 15u), 32);
}

// tanh forward-over-forward channel mixing applied to one 16x16 WMMA D-tile
// (channels x features), result written to the wave's LDS h-buffer.
// Channels: 0=primal, 1..4 = dz_mu, 5..14 = ddz_{mu,nu} pairs
// (0,0)(0,1)(0,2)(0,3)(1,1)(1,2)(1,3)(2,2)(2,3)(3,3), 15 = pad.
__device__ __forceinline__ void activate_store(float* c, float* hbuf, int nt,
                                               int col, bool lo,
                                               const float* __restrict__ bias) {
  float bval = bias[nt * 16 + col];          // all lanes load; select below
  c[0] += lo ? bval : 0.0f;                  // bias only on primal channel
  float z0 = bc16(c[0]);
  float z1 = bc16(c[1]);
  float z2 = bc16(c[2]);
  float z3 = bc16(c[3]);
  float z4 = bc16(c[4]);
  float h0 = fast_tanh(z0);
  float sx = 1.0f - h0 * h0;
  float g  = -2.0f * h0 * sx;
  // pair products for low-half channels (v -> ch v) and high-half (v -> ch v+8)
  float pl[8] = {0.f, 0.f, 0.f, 0.f, 0.f, z1 * z1, z1 * z2, z1 * z3};
  float ph[8] = {z1 * z4, z2 * z2, z2 * z3, z2 * z4,
                 z3 * z3, z3 * z4, z4 * z4, 0.f};
#pragma unroll
  for (int v = 0; v < 8; ++v) {
    float val = sx * c[v] + g * (lo ? pl[v] : ph[v]);
    if (v == 0 && lo)  val = h0;   // primal channel: tanh itself
    if (v == 7 && !lo) val = 0.f;  // pad channel 15
    int ch = v + (lo ? 0 : 8);
    hbuf[ch * H + nt * 16 + col] = val;
  }
}

// One wave == one sample: M=16 (AD channels), all three layers via
// V_WMMA_F32_16X16X4_F32. W2 is staged into LDS by the Tensor Data Mover.
// Writes e/de/dde flat (16 channels x 16) to ws.
__global__ __launch_bounds__(128) void tetrad_fwd_kernel(
    const float* __restrict__ x,  const float* __restrict__ W1,
    const float* __restrict__ b1, const float* __restrict__ W2,
    const float* __restrict__ b2, const float* __restrict__ W3,
    const float* __restrict__ b3, float* __restrict__ ws) {
  __shared__ float sW2[H * H];              // 64 KB: staged weights
  __shared__ float hb[4][16 * H];           // 32 KB: one h-buffer per wave
  const int  lane = (int)(threadIdx.x & 31u);
  const int  wave = (int)(threadIdx.x >> 5u);
  const int  col  = lane & 15;              // tile column / A row (M)
  const bool lo   = lane < 16;
  const int  k0   = lo ? 0 : 2;             // A/B K-pair layout: {0,1} lo, {2,3} hi
  float* hbuf = hb[wave];

  __builtin_prefetch(W3, 0, 0);             // W3 stays in global (L2-resident)

  // ---- stage W2 (128x128 f32) into LDS via the TDM (one descriptor, wave 0) ----
#if __has_builtin(__builtin_amdgcn_tensor_load_to_lds)
  if (wave == 0) {
    uint64_t ga  = (uint64_t)(uintptr_t)W2;
    unsigned lo32 = (unsigned)(ga & 0xffffffffull);
    unsigned hi25 = (unsigned)(ga >> 32);
    v4u g0;
    g0.x = 1u;                                  // count=1 valid user descriptor
    g0.y = (unsigned)(uintptr_t)sW2;            // lds_addr (bytes, addr[31:0])
    g0.z = lo32;                                // global_addr[31:0]
    g0.w = hi25 | (2u << 30);                   // global_addr[56:32] | type=2
    v8i g1;
    g1[0] = 0x00020000;                         // data_size=2 (4 bytes/elem)
    g1[1] = (H << 16);                          // tensor_dim0 = 128 (bits 63:48)
    g1[2] = (H << 16);                          // tensor_dim1 = 128 (bits 95:80)
    g1[3] = (H << 16);                          // tile_dim0   = 128 (bits 127:112)
    g1[4] = H;                                  // tile_dim1   = 128 (bits 143:128)
    g1[5] = H;                                  // tensor_dim0_stride = 128
    g1[6] = 0;
    g1[7] = 0;
    v4i gz = {0, 0, 0, 0};
#if __clang_major__ >= 23
    v8i gz8 = {0, 0, 0, 0, 0, 0, 0, 0};
    __builtin_amdgcn_tensor_load_to_lds(g0, g1, gz, gz, gz8, 0);
#else
    __builtin_amdgcn_tensor_load_to_lds(g0, g1, gz, gz, 0);
#endif
#if __has_builtin(__builtin_amdgcn_s_wait_tensorcnt)
    __builtin_amdgcn_s_wait_tensorcnt(0);
#endif
  }
  __syncthreads();
#else
  for (int i = (int)threadIdx.x; i < H * H; i += 128) sW2[i] = W2[i];
  __syncthreads();
#endif

  const int wglobal = blockIdx.x * 4 + wave;
  const int wstride = gridDim.x * 4;

  for (int s = wglobal; s < NB; s += wstride) {
    const float* xs = x + (size_t)s * 4;

    // ---------- layer 1: A(16x4) synthesized in regs, B = W1 (4x128) ----------
    // row 0 = x, rows 1..4 = unit tangent directions, rows 5..15 = 0
    v2f a1;
    {
      float x0 = xs[0], x1 = xs[1], x2 = xs[2], x3 = xs[3];  // branchless loads
      float xa = lo ? x0 : x2;   // xs[k0]
      float xb = lo ? x1 : x3;   // xs[k0+1]
      float u0 = (col >= 1 && col <= 4 && k0 == col - 1) ? 1.f : 0.f;
      float u1 = (col >= 1 && col <= 4 && (k0 + 1) == col - 1) ? 1.f : 0.f;
      a1.x = (col == 0) ? xa : u0;
      a1.y = (col == 0) ? xb : u1;
    }
    v8f acc[8];
#pragma unroll
    for (int nt = 0; nt < 8; ++nt) {
      v2f bv;
      bv.x = W1[k0 * H + nt * 16 + col];
      bv.y = W1[(k0 + 1) * H + nt * 16 + col];
      v8f cz = {};
      acc[nt] = __builtin_amdgcn_wmma_f32_16x16x4_f32(false, a1, false, bv,
                                                      (short)0, cz, false, false);
    }
#pragma unroll
    for (int nt = 0; nt < 8; ++nt)
      activate_store((float*)&acc[nt], hbuf, nt, col, lo, b1);

    // ---------- layer 2: [16xH] @ [HxH], B from LDS-staged W2 ----------
    {
      v8f cz = {};
#pragma unroll
      for (int nt = 0; nt < 8; ++nt) acc[nt] = cz;
    }
    for (int kk = 0; kk < H / 4; ++kk) {
      v2f av = *(const v2f*)&hbuf[col * H + 4 * kk + k0];  // ds_load_b64
#pragma unroll
      for (int nt = 0; nt < 8; ++nt) {
        v2f bv;
        bv.x = sW2[(4 * kk + k0) * H + nt * 16 + col];
        bv.y = sW2[(4 * kk + k0 + 1) * H + nt * 16 + col];
        acc[nt] = __builtin_amdgcn_wmma_f32_16x16x4_f32(false, av, false, bv,
                                                        (short)0, acc[nt], false, false);
      }
    }
#pragma unroll
    for (int nt = 0; nt < 8; ++nt)
      activate_store((float*)&acc[nt], hbuf, nt, col, lo, b2);

    // ---------- layer 3: [16xH] @ [Hx16] (linear output) ----------
    v8f c3 = {};
    for (int kk = 0; kk < H / 4; ++kk) {
      v2f av = *(const v2f*)&hbuf[col * H + 4 * kk + k0];
      v2f bv;
      bv.x = W3[(4 * kk + k0) * 16 + col];
      bv.y = W3[(4 * kk + k0 + 1) * 16 + col];
      c3 = __builtin_amdgcn_wmma_f32_16x16x4_f32(false, av, false, bv,
                                                 (short)0, c3, false, false);
    }
    float* cf = (float*)&c3;
    float b3v = b3[col];
    cf[0] += lo ? b3v : 0.0f;
    float* wq = ws + (size_t)s * 256;
#pragma unroll
    for (int v = 0; v < 8; ++v) {
      int ch = v + (lo ? 0 : 8);
      wq[ch * 16 + col] = cf[v];
    }
  }
}

// ---------------------------------------------------------------------------
// 4x4 inverse via adjugate
__device__ void inv4(const float* m, float* inv) {
  inv[0]  =  m[5]*m[10]*m[15] - m[5]*m[11]*m[14] - m[9]*m[6]*m[15] + m[9]*m[7]*m[14] + m[13]*m[6]*m[11] - m[13]*m[7]*m[10];
  inv[4]  = -m[4]*m[10]*m[15] + m[4]*m[11]*m[14] + m[8]*m[6]*m[15] - m[8]*m[7]*m[14] - m[12]*m[6]*m[11] + m[12]*m[7]*m[10];
  inv[8]  =  m[4]*m[9]*m[15]  - m[4]*m[11]*m[13] - m[8]*m[5]*m[15] + m[8]*m[7]*m[13] + m[12]*m[5]*m[11] - m[12]*m[7]*m[9];
  inv[12] = -m[4]*m[9]*m[14]  + m[4]*m[10]*m[13] + m[8]*m[5]*m[14] - m[8]*m[6]*m[13] - m[12]*m[5]*m[10] + m[12]*m[6]*m[9];
  inv[1]  = -m[1]*m[10]*m[15] + m[1]*m[11]*m[14] + m[9]*m[2]*m[15] - m[9]*m[3]*m[14] - m[13]*m[2]*m[11] + m[13]*m[3]*m[10];
  inv[5]  =  m[0]*m[10]*m[15] - m[0]*m[11]*m[14] - m[8]*m[2]*m[15] + m[8]*m[3]*m[14] + m[12]*m[2]*m[11] - m[12]*m[3]*m[10];
  inv[9]  = -m[0]*m[9]*m[15]  + m[0]*m[11]*m[13] + m[8]*m[1]*m[15] - m[8]*m[3]*m[13] - m[12]*m[1]*m[11] + m[12]*m[3]*m[9];
  inv[13] =  m[0]*m[9]*m[14]  - m[0]*m[10]*m[13] - m[8]*m[1]*m[14] + m[8]*m[2]*m[13] + m[12]*m[1]*m[10] - m[12]*m[2]*m[9];
  inv[2]  =  m[1]*m[6]*m[15]  - m[1]*m[7]*m[14]  - m[5]*m[2]*m[15] + m[5]*m[3]*m[14] + m[13]*m[2]*m[7]  - m[13]*m[3]*m[6];
  inv[6]  = -m[0]*m[6]*m[15]  + m[0]*m[7]*m[14]  + m[4]*m[2]*m[15] - m[4]*m[3]*m[14] - m[12]*m[2]*m[7]  + m[12]*m[3]*m[6];
  inv[10] =  m[0]*m[5]*m[15]  - m[0]*m[7]*m[13]  - m[4]*m[1]*m[15] + m[4]*m[3]*m[13] + m[12]*m[1]*m[7]  - m[12]*m[3]*m[5];
  inv[14] = -m[0]*m[5]*m[14]  + m[0]*m[6]*m[13]  + m[4]*m[1]*m[14] - m[4]*m[2]*m[13] - m[12]*m[1]*m[6]  + m[12]*m[2]*m[5];
  inv[3]  = -m[1]*m[6]*m[11]  + m[1]*m[7]*m[10]  + m[5]*m[2]*m[11] - m[5]*m[3]*m[10] - m[9]*m[2]*m[7]   + m[9]*m[3]*m[6];
  inv[7]  =  m[0]*m[6]*m[11]  - m[0]*m[7]*m[10]  - m[4]*m[2]*m[11] + m[4]*m[3]*m[10] + m[8]*m[2]*m[7]   - m[8]*m[3]*m[6];
  inv[11] = -m[0]*m[5]*m[11]  + m[0]*m[7]*m[9]   + m[4]*m[1]*m[11] - m[4]*m[3]*m[9]  - m[8]*m[1]*m[7]   + m[8]*m[3]*m[5];
  inv[15] =  m[0]*m[5]*m[10]  - m[0]*m[6]*m[9]   - m[4]*m[1]*m[10] + m[4]*m[2]*m[9]  + m[8]*m[1]*m[6]   - m[8]*m[2]*m[5];
  float det = m[0]*inv[0] + m[1]*inv[4] + m[2]*inv[8] + m[3]*inv[12];
  float r = 1.0f / det;
  for (int i = 0; i < 16; ++i) inv[i] *= r;
}

// Per-sample 4x4 tensor algebra epilogue. One thread per sample.
__global__ __launch_bounds__(256) void einstein_epilogue(
    const float* __restrict__ ws, float* __restrict__ out) {
  int s = blockIdx.x * blockDim.x + threadIdx.x;
  if (s >= NB) return;
  const float* q = ws + (size_t)s * 256;
  const float eta[4] = {-1.f, 1.f, 1.f, 1.f};
  const int P[4][4] = {{0,1,2,3},{1,4,5,6},{2,5,7,8},{3,6,8,9}}; // sym pair index

  float e[16], de[4][16], dd[10][16];
  for (int i = 0; i < 16; ++i) e[i] = q[i];
  for (int u = 0; u < 4; ++u)
    for (int i = 0; i < 16; ++i) de[u][i] = q[(1 + u) * 16 + i];
  for (int p = 0; p < 10; ++p)
    for (int i = 0; i < 16; ++i) dd[p][i] = q[(5 + p) * 16 + i];

  float ei[16];
  inv4(e, ei);                               // ei[I*4+m]
  float em[16];                              // e_mat[m][I] = eta_I * einv[I][m]
  for (int mm = 0; mm < 4; ++mm)
    for (int I = 0; I < 4; ++I) em[mm*4+I] = eta[I] * ei[I*4+mm];

  // d(e_mat)_u = eta_I * ( -einv * dE_u * einv )[I][m]
  float dem[4][16];
  for (int u = 0; u < 4; ++u) {
    float t[16];
    for (int I = 0; I < 4; ++I)
      for (int b = 0; b < 4; ++b) {
        float a = 0.f;
        for (int aa = 0; aa < 4; ++aa) a += ei[I*4+aa] * de[u][aa*4+b];
        t[I*4+b] = a;
      }
    for (int I = 0; I < 4; ++I)
      for (int mp = 0; mp < 4; ++mp) {
        float a = 0.f;
        for (int b = 0; b < 4; ++b) a += t[I*4+b] * ei[b*4+mp];
        dem[u][mp*4+I] = -eta[I] * a;
      }
  }

  // rot[I][J][K] = sum_{m,n} em[m][I] em[n][J] de[m][n][K]
  float rot[64];
  for (int I = 0; I < 4; ++I)
    for (int J = 0; J < 4; ++J)
      for (int K = 0; K < 4; ++K) {
        float a = 0.f;
        for (int mi = 0; mi < 4; ++mi)
          for (int n = 0; n < 4; ++n)
            a += em[mi*4+I] * em[n*4+J] * de[mi][n*4+K];
        rot[(I*4+J)*4+K] = a;
      }
  float r2[64];
  {
    float tau[64];
    for (int I = 0; I < 4; ++I)
      for (int J = 0; J < 4; ++J)
        for (int K = 0; K < 4; ++K)
          tau[(I*4+J)*4+K] = rot[(I*4+J)*4+K] - rot[(J*4+I)*4+K];
    for (int I = 0; I < 4; ++I)
      for (int J = 0; J < 4; ++J)
        for (int K = 0; K < 4; ++K)
          r2[(I*4+J)*4+K] = 0.5f * (tau[(K*4+J)*4+I] + tau[(I*4+J)*4+K] - tau[(I*4+K)*4+J]);
  }

  // w[m][I][J] = sum_K r2[K][I][J] e[m][K] eta_K ;  w2 = w * eta_J
  float w[64], w2[64];
  for (int mi = 0; mi < 4; ++mi)
    for (int I = 0; I < 4; ++I)
      for (int J = 0; J < 4; ++J) {
        float a = 0.f;
        for (int K = 0; K < 4; ++K) a += r2[(K*4+I)*4+J] * e[mi*4+K] * eta[K];
        w[(mi*4+I)*4+J]  = a;
        w2[(mi*4+I)*4+J] = a * eta[J];
      }

  // dw2[u][m][I][J]
  float dw2[4][64];
  for (int u = 0; u < 4; ++u) {
    float dr[64];
    for (int I = 0; I < 4; ++I)
      for (int J = 0; J < 4; ++J)
        for (int K = 0; K < 4; ++K) {
          float a = 0.f;
          for (int mi = 0; mi < 4; ++mi)
            for (int n = 0; n < 4; ++n)
              a += (dem[u][mi*4+I] * em[n*4+J] + em[mi*4+I] * dem[u][n*4+J]) * de[mi][n*4+K]
                 + em[mi*4+I] * em[n*4+J] * dd[P[u][mi]][n*4+K];
          dr[(I*4+J)*4+K] = a;
        }
    float dt[64], dr2[64];
    for (int I = 0; I < 4; ++I)
      for (int J = 0; J < 4; ++J)
        for (int K = 0; K < 4; ++K)
          dt[(I*4+J)*4+K] = dr[(I*4+J)*4+K] - dr[(J*4+I)*4+K];
    for (int I = 0; I < 4; ++I)
      for (int J = 0; J < 4; ++J)
        for (int K = 0; K < 4; ++K)
          dr2[(I*4+J)*4+K] = 0.5f * (dt[(K*4+J)*4+I] + dt[(I*4+J)*4+K] - dt[(I*4+K)*4+J]);
    for (int mi = 0; mi < 4; ++mi)
      for (int I = 0; I < 4; ++I)
        for (int J = 0; J < 4; ++J) {
          float a = 0.f;
          for (int K = 0; K < 4; ++K)
            a += eta[K] * (dr2[(K*4+I)*4+J] * e[mi*4+K] + r2[(K*4+I)*4+J] * de[u][mi*4+K]);
          dw2[u][(mi*4+I)*4+J] = a * eta[J];
        }
  }

  // Riemann + antisymmetrize m<->n
  float ra[256];
  {
    float rie[256];
    for (int mi = 0; mi < 4; ++mi)
      for (int n = 0; n < 4; ++n)
        for (int I = 0; I < 4; ++I)
          for (int J = 0; J < 4; ++J) {
            float a = dw2[mi][(n*4+I)*4+J];
            for (int k = 0; k < 4; ++k)
              a += w2[(mi*4+I)*4+k] * w2[(n*4+k)*4+J];
            rie[((mi*4+n)*4+I)*4+J] = a;
          }
    for (int mi = 0; mi < 4; ++mi)
      for (int n = 0; n < 4; ++n)
        for (int I = 0; I < 4; ++I)
          for (int J = 0; J < 4; ++J)
            ra[((mi*4+n)*4+I)*4+J] = rie[((mi*4+n)*4+I)*4+J] - rie[((n*4+mi)*4+I)*4+J];
  }

  // Ricci
  float* oric = out + (size_t)s * 16;
  {
    float r1[16];
    for (int mi = 0; mi < 4; ++mi)
      for (int I = 0; I < 4; ++I) {
        float a = 0.f;
        for (int n = 0; n < 4; ++n)
          for (int j = 0; j < 4; ++j)
            a += ra[((mi*4+n)*4+I)*4+j] * em[n*4+j];
        r1[mi*4+I] = a;
      }
    for (int i = 0; i < 4; ++i)
      for (int j = 0; j < 4; ++j) {
        float a = 0.f;
        for (int mi = 0; mi < 4; ++mi) a += r1[mi*4+j] * em[mi*4+i];
        oric[i*4+j] = a;
      }
  }

  // riem_out[K][L][I][M] = eta_M * sum_{m,n} ra[m][n][I][M] em[m][K] em[n][L]
  float* oro = out + (size_t)NB * 16 + (size_t)s * 256;
  {
    float U[256];
    for (int K = 0; K < 4; ++K)
      for (int n = 0; n < 4; ++n)
        for (int I = 0; I < 4; ++I)
          for (int M = 0; M < 4; ++M) {
            float a = 0.f;
            for (int mi = 0; mi < 4; ++mi)
              a += ra[((mi*4+n)*4+I)*4+M] * em[mi*4+K];
            U[((K*4+n)*4+I)*4+M] = a;
          }
    for (int K = 0; K < 4; ++K)
      for (int L = 0; L < 4; ++L)
        for (int I = 0; I < 4; ++I)
          for (int M = 0; M < 4; ++M) {
            float a = 0.f;
            for (int n = 0; n < 4; ++n)
              a += U[((K*4+n)*4+I)*4+M] * em[n*4+L];
            oro[((K*4+L)*4+I)*4+M] = eta[M] * a;
          }
  }

  // wald_1 = de - de^T(m<->n)
  float* ow1 = out + (size_t)NB * 272 + (size_t)s * 64;
  for (int mi = 0; mi < 4; ++mi)
    for (int n = 0; n < 4; ++n)
      for (int I = 0; I < 4; ++I)
        ow1[(mi*4+n)*4+I] = de[mi][n*4+I] - de[n][mi*4+I];

  // wald_2 = e.w2 antisymmetrized
  float* ow2 = out + (size_t)NB * 336 + (size_t)s * 64;
  {
    float wr[64];
    for (int mi = 0; mi < 4; ++mi)
      for (int n = 0; n < 4; ++n)
        for (int I = 0; I < 4; ++I) {
          float a = 0.f;
          for (int J = 0; J < 4; ++J) a += e[mi*4+J] * w2[(n*4+I)*4+J];
          wr[(mi*4+n)*4+I] = a;
        }
    for (int mi = 0; mi < 4; ++mi)
      for (int n = 0; n < 4; ++n)
        for (int I = 0; I < 4; ++I)
          ow2[(mi*4+n)*4+I] = wr[(mi*4+n)*4+I] - wr[(n*4+mi)*4+I];
  }

  float* ow = out + (size_t)NB * 400 + (size_t)s * 64;
  for (int i = 0; i < 64; ++i) ow[i] = w[i];
}

extern "C" void kernel_launch(void* const* d_in, const int* in_sizes, int n_in,
                              void* d_out, int out_size, void* d_ws, size_t ws_size,
                              hipStream_t stream) {
  const float* x  = (const float*)d_in[0];
  const float* W1 = (const float*)d_in[1];
  const float* b1 = (const float*)d_in[2];
  const float* W2 = (const float*)d_in[3];
  const float* b2 = (const float*)d_in[4];
  const float* W3 = (const float*)d_in[5];
  const float* b3 = (const float*)d_in[6];
  float* wsF = (float*)d_ws;   // needs 8192 * 256 * 4 = 8 MB
  float* out = (float*)d_out;

  // 2048 blocks x 4 waves = 8192 waves -> one sample per wave
  tetrad_fwd_kernel<<<dim3(2048), dim3(128), 0, stream>>>(x, W1, b1, W2, b2, W3, b3, wsF);
  einstein_epilogue<<<dim3(NB / 256), dim3(256), 0, stream>>>(wsF, out);
}